// EpisodicSlotReader_12283606467870
// MI455X (gfx1250) — compile-verified
//
#include <hip/hip_runtime.h>
#include <math.h>

// Problem constants (reference: B=2048, S=256, D=512, fp32 everywhere)
#define B_N 2048
#define S_N 256
#define D_N 512

#define NTHREADS 256
#define NWAVES 8
#define ROWS_PER_WAVE 32          // 8 waves * 32 rows = 256 = S
#define CHUNK 16                  // columns staged per chunk
#define NCHUNK (D_N / CHUNK)      // 32 chunks
#define KSTRIDE (CHUNK + 4)       // 20 dwords: 16B-aligned rows + conflict-free banks

typedef float v2f __attribute__((ext_vector_type(2)));
typedef float v4f __attribute__((ext_vector_type(4)));
typedef float v8f __attribute__((ext_vector_type(8)));
typedef int i32x4v __attribute__((vector_size(16)));  // matches builtin param type

// ---------------- CDNA5 feature probes (compile-loop guided) ----------------
#if __has_builtin(__builtin_amdgcn_wmma_f32_16x16x4_f32)
#define HAVE_WMMA_F32X4 1
#else
#define HAVE_WMMA_F32X4 0
#endif

#if __has_builtin(__builtin_amdgcn_global_load_async_to_lds_b128)
#define HAVE_ASYNC 1
__device__ __forceinline__ void async_cp_b128(const float* g, float* l) {
  // param0: v4i addrspace(1)*  (clang prints AS1 as "__device__")
  // param1: v4i addrspace(3)*
  __builtin_amdgcn_global_load_async_to_lds_b128(
      (__attribute__((address_space(1))) i32x4v*)g,
      (__attribute__((address_space(3))) i32x4v*)l, 0, 0);
}
#else
#define HAVE_ASYNC 0
#endif

#if HAVE_ASYNC
#if __has_builtin(__builtin_amdgcn_s_wait_asynccnt)
#define WAIT_ASYNC(n) do { __builtin_amdgcn_s_wait_asynccnt(n); asm volatile("" ::: "memory"); } while (0)
#else
#define WAIT_ASYNC(n) asm volatile("s_wait_asynccnt %0" :: "n"(n) : "memory")
#endif
#else
// fallback staging is plain load+ds_store; just a compiler reordering fence
#define WAIT_ASYNC(n) asm volatile("" ::: "memory")
#endif

// ---------------- block reductions (wave32, 8 waves) ----------------
__device__ __forceinline__ float block_sum(float v, float* red) {
#pragma unroll
  for (int m = 16; m > 0; m >>= 1) v += __shfl_xor(v, m);
  const int wv = threadIdx.x >> 5, ln = threadIdx.x & 31;
  if (ln == 0) red[wv] = v;
  __syncthreads();
  float t = 0.f;
#pragma unroll
  for (int i = 0; i < NWAVES; ++i) t += red[i];
  __syncthreads();
  return t;
}

__device__ __forceinline__ float block_max(float v, float* red) {
#pragma unroll
  for (int m = 16; m > 0; m >>= 1) v = fmaxf(v, __shfl_xor(v, m));
  const int wv = threadIdx.x >> 5, ln = threadIdx.x & 31;
  if (ln == 0) red[wv] = v;
  __syncthreads();
  float t = red[0];
#pragma unroll
  for (int i = 1; i < NWAVES; ++i) t = fmaxf(t, red[i]);
  __syncthreads();
  return t;
}

// ---------------- kernel: one block per batch row b ----------------
__global__ __launch_bounds__(NTHREADS) void episodic_slot_reader_kernel(
    const float* __restrict__ Q, const float* __restrict__ K,
    const float* __restrict__ V, const float* __restrict__ AGE,
    const float* __restrict__ STR, const float* __restrict__ SCALE,
    float* __restrict__ out_read, float* __restrict__ out_w,
    float* __restrict__ out_logits) {
  __shared__ __align__(16) float lds_qn[D_N];
  __shared__ __align__(16) float lds_k[NWAVES][2][ROWS_PER_WAVE * KSTRIDE];
  __shared__ float lds_dot[S_N];
  __shared__ float lds_ss[S_N];
  __shared__ float lds_w[S_N];
  __shared__ float lds_red[NWAVES];

  const int b    = blockIdx.x;
  const int tid  = threadIdx.x;
  const int wave = tid >> 5;
  const int lane = tid & 31;

  const float* Kb = K + (size_t)b * S_N * D_N;
  const float* Vb = V + (size_t)b * S_N * D_N;
  const int s_base = wave * ROWS_PER_WAVE;

  // Staging geometry: one b128 instr moves 8 rows x 64B (16 floats); 4 instrs = 32x16 tile.
  const int st_row = lane >> 2;        // 0..7
  const int st_col = (lane & 3) * 4;   // 0,4,8,12 floats

  auto issue_chunk = [&](int bufi, int ch) {
    const int d0 = ch * CHUNK;
    float* lbase = &lds_k[wave][bufi][0];
    const float* gbase = Kb + (size_t)s_base * D_N + d0;
#pragma unroll
    for (int j = 0; j < 4; ++j) {
      const int row = j * 8 + st_row;
      const float* g = gbase + (size_t)row * D_N + st_col;
      float* l = lbase + row * KSTRIDE + st_col;
#if HAVE_ASYNC
      async_cp_b128(g, l);
#else
      v4f val = __builtin_nontemporal_load((const v4f*)g);
      *(v4f*)l = val;
#endif
    }
  };

  // ---- Phase 0: q load + prefetch of first K chunk + q L2-normalize ----
  const float q0v = Q[(size_t)b * D_N + tid];
  const float q1v = Q[(size_t)b * D_N + tid + 256];
  issue_chunk(0, 0);  // overlap HBM latency with the q reduction
  const float qss = block_sum(q0v * q0v + q1v * q1v, lds_red);
  const float qinv = 1.f / (sqrtf(qss) + 1e-6f);
  lds_qn[tid] = q0v * qinv;
  lds_qn[tid + 256] = q1v * qinv;
  __syncthreads();

  // ---- Phase 1: sim = K_row . qn via WMMA f32 16x16x4; ||K_row||^2 on VALU ----
  v8f c0 = {};
  v8f c1 = {};
  float ss0 = 0.f, ss1 = 0.f;
#if !HAVE_WMMA_F32X4
  float dp0 = 0.f, dp1 = 0.f;
#endif

  for (int ch = 0; ch < NCHUNK; ++ch) {
    if (ch + 1 < NCHUNK) {
      issue_chunk((ch + 1) & 1, ch + 1);
      WAIT_ASYNC(4);   // previous chunk (4 async ops) complete; next 4 in flight
    } else {
      WAIT_ASYNC(0);
    }
    const float* buf = &lds_k[wave][ch & 1][0];
    const float* qch = &lds_qn[ch * CHUNK];
#pragma unroll
    for (int dk = 0; dk < CHUNK; dk += 4) {
      // ISA A-layout 16x4 f32: lanes 0-15 -> cols {k,k+1}; lanes 16-31 -> {k+2,k+3}
      const int coff = dk + ((lane >> 4) << 1);
      const v2f bfrag = *(const v2f*)(qch + coff);  // qn broadcast over all N cols
      const v2f a0 = *(const v2f*)(buf + (lane & 15) * KSTRIDE + coff);
      const v2f a1 = *(const v2f*)(buf + ((lane & 15) + 16) * KSTRIDE + coff);
      ss0 += a0.x * a0.x + a0.y * a0.y;
      ss1 += a1.x * a1.x + a1.y * a1.y;
#if HAVE_WMMA_F32X4
      c0 = __builtin_amdgcn_wmma_f32_16x16x4_f32(false, a0, false, bfrag, (short)0, c0, false, false);
      c1 = __builtin_amdgcn_wmma_f32_16x16x4_f32(false, a1, false, bfrag, (short)0, c1, false, false);
#else
      dp0 += a0.x * bfrag.x + a0.y * bfrag.y;
      dp1 += a1.x * bfrag.x + a1.y * bfrag.y;
#endif
    }
  }

  // row sums: lane L and L+16 hold complementary column halves of row (L&15)
  ss0 += __shfl_xor(ss0, 16);
  ss1 += __shfl_xor(ss1, 16);
#if HAVE_WMMA_F32X4
  // C layout: VGPR r: lanes 0-15 -> M=r, lanes 16-31 -> M=8+r (all N cols equal)
  if (lane == 0) {
#pragma unroll
    for (int r = 0; r < 8; ++r) {
      lds_dot[s_base + r] = c0[r];
      lds_dot[s_base + 16 + r] = c1[r];
    }
  }
  if (lane == 16) {
#pragma unroll
    for (int r = 0; r < 8; ++r) {
      lds_dot[s_base + 8 + r] = c0[r];
      lds_dot[s_base + 24 + r] = c1[r];
    }
  }
#else
  dp0 += __shfl_xor(dp0, 16);
  dp1 += __shfl_xor(dp1, 16);
  if (lane < 16) {
    lds_dot[s_base + lane] = dp0;
    lds_dot[s_base + 16 + lane] = dp1;
  }
#endif
  if (lane < 16) {
    lds_ss[s_base + lane] = ss0;
    lds_ss[s_base + 16 + lane] = ss1;
  }
  (void)c0; (void)c1;
  __syncthreads();

  // ---- Phase 2: logits + softmax over S (thread t <-> slot t) ----
  const size_t bs = (size_t)b * S_N;
  const float strv = STR[bs + tid];
  const float agev = AGE[bs + tid];
  const float sim = lds_dot[tid] / (sqrtf(lds_ss[tid]) + 1e-6f);
  float lg = sim + 0.5f * logf(fminf(fmaxf(strv, 1e-3f), 1e9f)) - 0.02f * agev;
  lg += (strv > 1e-3f) ? 0.f : -1000.f;
  out_logits[bs + tid] = lg;

  const float mx = block_max(lg, lds_red);
  const float ex = expf(lg - mx);
  const float sm = block_sum(ex, lds_red);
  const float wgt = ex / sm;
  out_w[bs + tid] = wgt;
  lds_w[tid] = wgt;
  __syncthreads();

  // ---- Phase 3: read = sum_s w[s] * V[b,s,:]; thread owns cols {t, t+256} ----
  float acc0 = 0.f, acc1 = 0.f;
#pragma unroll 4
  for (int s = 0; s < S_N; ++s) {
    const float wv = lds_w[s];
    acc0 += wv * __builtin_nontemporal_load(Vb + (size_t)s * D_N + tid);
    acc1 += wv * __builtin_nontemporal_load(Vb + (size_t)s * D_N + tid + 256);
  }

  // RMSNorm + scale
  const float rss = block_sum(acc0 * acc0 + acc1 * acc1, lds_red);
  const float rinv = 1.f / sqrtf(rss * (1.f / (float)D_N) + 1e-6f);
  out_read[(size_t)b * D_N + tid] = acc0 * rinv * SCALE[tid];
  out_read[(size_t)b * D_N + tid + 256] = acc1 * rinv * SCALE[tid + 256];
}

extern "C" void kernel_launch(void* const* d_in, const int* in_sizes, int n_in,
                              void* d_out, int out_size, void* d_ws, size_t ws_size,
                              hipStream_t stream) {
  (void)in_sizes; (void)n_in; (void)out_size; (void)d_ws; (void)ws_size;
  const float* q    = (const float*)d_in[0];  // [B, D]
  const float* keys = (const float*)d_in[1];  // [B, S, D]
  const float* vals = (const float*)d_in[2];  // [B, S, D]
  const float* age  = (const float*)d_in[3];  // [B, S]
  const float* str  = (const float*)d_in[4];  // [B, S]
  const float* scl  = (const float*)d_in[5];  // [D]
  float* out        = (float*)d_out;          // read | w | logits, flat
  float* out_read   = out;
  float* out_w      = out + (size_t)B_N * D_N;
  float* out_logits = out_w + (size_t)B_N * S_N;

  episodic_slot_reader_kernel<<<dim3(B_N), dim3(NTHREADS), 0, stream>>>(
      q, keys, vals, age, str, scl, out_read, out_w, out_logits);
}